// GraphSAGE_21251498181087
// MI455X (gfx1250) — compile-verified
//
#include <hip/hip_runtime.h>
#include <hip/hip_bf16.h>

#define D_FEAT 128

typedef __attribute__((ext_vector_type(2))) float v2f;
typedef __attribute__((ext_vector_type(8))) float v8f;

// ---------------------------------------------------------------------------
// Zero a float buffer
// ---------------------------------------------------------------------------
__global__ __launch_bounds__(256) void k_zero(float* __restrict__ p, int count) {
    int i = blockIdx.x * blockDim.x + threadIdx.x;
    if (i < count) p[i] = 0.0f;
}

// ---------------------------------------------------------------------------
// Degree: atomicAdd 1.0 per edge into deg[dst]
// ---------------------------------------------------------------------------
__global__ __launch_bounds__(256) void k_degree(const int* __restrict__ dst,
                                                float* __restrict__ deg, int n_edges) {
    int i = blockIdx.x * blockDim.x + threadIdx.x;
    if (i < n_edges) atomicAdd(&deg[dst[i]], 1.0f);
}

// ---------------------------------------------------------------------------
// inv[i] = 1 / max(deg[i], 1)
// ---------------------------------------------------------------------------
__global__ __launch_bounds__(256) void k_invdeg(const float* __restrict__ deg,
                                                float* __restrict__ inv, int n) {
    int i = blockIdx.x * blockDim.x + threadIdx.x;
    if (i < n) inv[i] = 1.0f / fmaxf(deg[i], 1.0f);
}

// ---------------------------------------------------------------------------
// Edge scatter: one wave32 per edge. Each lane carries 4 floats (32*4 = 128 = D).
// nsum[dst] += h[src]
// ---------------------------------------------------------------------------
__global__ __launch_bounds__(256) void k_scatter(const float* __restrict__ h,
                                                 const int* __restrict__ src,
                                                 const int* __restrict__ dst,
                                                 float* __restrict__ nsum, int n_edges) {
    int gtid = blockIdx.x * blockDim.x + threadIdx.x;
    int edge = gtid >> 5;
    int lane = gtid & 31;
    if (edge >= n_edges) return;
    int s = src[edge];
    int d = dst[edge];
    const float4 v = *(const float4*)(h + (size_t)s * D_FEAT + lane * 4);
    float* o = nsum + (size_t)d * D_FEAT + lane * 4;
    atomicAdd(o + 0, v.x);
    atomicAdd(o + 1, v.y);
    atomicAdd(o + 2, v.z);
    atomicAdd(o + 3, v.w);
}

// ---------------------------------------------------------------------------
// Fused SAGE layer GEMM using V_WMMA_F32_16X16X4_F32 (fp32, full precision):
//   out[m, n] = hs[m, :] @ Wself[n, :] + (nsum[m, :] * inv[m]) @ Wneigh[n, :] + b[n]
// One wave computes one 16x16 output tile. Block = (32, 8): 16 nodes x 128 outputs.
//
// A layout (32-bit, 16x4, 2 VGPRs):  lanes 0-15: M=lane,  K = {kk, kk+1}
//                                    lanes 16-31: M=lane-16, K = {kk+2, kk+3}
// B layout mirrors A with lane = N column index; B[k, n] = W[n, k].
// C/D layout: VGPR r -> row = r + 8*(lane>=16), col = lane & 15.
// ---------------------------------------------------------------------------
__global__ __launch_bounds__(256) void k_sage_gemm(const float* __restrict__ hs,
                                                   const float* __restrict__ nsum,
                                                   const float* __restrict__ invdeg,
                                                   const float* __restrict__ Wself,
                                                   const float* __restrict__ Wneigh,
                                                   const float* __restrict__ bias,
                                                   float* __restrict__ out,
                                                   int n_nodes, int do_relu) {
    const int lane  = threadIdx.x;        // 0..31
    const int ntile = threadIdx.y;        // 0..7  -> output-column block
    const int mbase = blockIdx.x * 16;
    const int half  = lane >> 4;          // 0 or 1
    const int l16   = lane & 15;

    // Clamp load row so EXEC stays all-1s (WMMA requirement); stores are guarded.
    int m = mbase + l16;
    if (m >= n_nodes) m = n_nodes - 1;

    const int ncol = ntile * 16 + l16;    // output column (= row of W)
    const int koff = half * 2;            // K offset within the group of 4

    const float inv = invdeg[m];
    const float* hs_row = hs   + (size_t)m * D_FEAT + koff;
    const float* ns_row = nsum + (size_t)m * D_FEAT + koff;
    const float* ws_row = Wself  + (size_t)ncol * D_FEAT + koff;
    const float* wn_row = Wneigh + (size_t)ncol * D_FEAT + koff;

    v8f acc = {};
    #pragma unroll
    for (int kk = 0; kk < D_FEAT; kk += 4) {
        v2f a_s = *(const v2f*)(hs_row + kk);
        v2f a_n = *(const v2f*)(ns_row + kk);
        a_n[0] *= inv;
        a_n[1] *= inv;
        v2f b_s = *(const v2f*)(ws_row + kk);
        v2f b_n = *(const v2f*)(wn_row + kk);
        acc = __builtin_amdgcn_wmma_f32_16x16x4_f32(
            false, a_s, false, b_s, (short)0, acc, false, false);
        acc = __builtin_amdgcn_wmma_f32_16x16x4_f32(
            false, a_n, false, b_n, (short)0, acc, false, false);
    }

    const float b = bias[ncol];
    #pragma unroll
    for (int r = 0; r < 8; ++r) {
        int row = mbase + r + 8 * half;
        if (row < n_nodes) {
            float v = acc[r] + b;
            if (do_relu) v = fmaxf(v, 0.0f);
            out[(size_t)row * D_FEAT + ncol] = v;
        }
    }
}

// ---------------------------------------------------------------------------
// Host-side launch
// ---------------------------------------------------------------------------
extern "C" void kernel_launch(void* const* d_in, const int* in_sizes, int n_in,
                              void* d_out, int out_size, void* d_ws, size_t ws_size,
                              hipStream_t stream) {
    const float* x        = (const float*)d_in[0];
    const int*   src      = (const int*)  d_in[1];
    const int*   dst      = (const int*)  d_in[2];
    const float* w_self1  = (const float*)d_in[3];
    const float* w_neigh1 = (const float*)d_in[4];
    const float* b1       = (const float*)d_in[5];
    const float* w_self2  = (const float*)d_in[6];
    const float* w_neigh2 = (const float*)d_in[7];
    const float* b2       = (const float*)d_in[8];
    float*       out      = (float*)d_out;

    const int n_nodes = in_sizes[0] / D_FEAT;   // 50000
    const int n_edges = in_sizes[1];            // 800000
    const size_t nd = (size_t)n_nodes * D_FEAT;

    float* ws   = (float*)d_ws;
    float* nsum = ws;                 // [N, D]
    float* h1   = ws + nd;            // [N, D]
    float* deg  = ws + 2 * nd;        // [N]
    float* inv  = deg + n_nodes;      // [N]

    const int T = 256;
    dim3 gblk((unsigned)((n_nodes + 15) / 16));
    dim3 gthr(32, 8);

    // ---- degree (shared by both layers) ----
    k_zero<<<(int)((nd + T - 1) / T), T, 0, stream>>>(nsum, (int)nd);
    k_zero<<<(n_nodes + T - 1) / T, T, 0, stream>>>(deg, n_nodes);
    k_degree<<<(n_edges + T - 1) / T, T, 0, stream>>>(dst, deg, n_edges);
    k_invdeg<<<(n_nodes + T - 1) / T, T, 0, stream>>>(deg, inv, n_nodes);

    // ---- layer 1 ----
    {
        long long waves = (long long)n_edges;
        int blocks = (int)((waves * 32 + T - 1) / T);
        k_scatter<<<blocks, T, 0, stream>>>(x, src, dst, nsum, n_edges);
    }
    k_sage_gemm<<<gblk, gthr, 0, stream>>>(x, nsum, inv, w_self1, w_neigh1, b1,
                                           h1, n_nodes, /*relu=*/1);

    // ---- layer 2 ----
    k_zero<<<(int)((nd + T - 1) / T), T, 0, stream>>>(nsum, (int)nd);
    {
        long long waves = (long long)n_edges;
        int blocks = (int)((waves * 32 + T - 1) / T);
        k_scatter<<<blocks, T, 0, stream>>>(h1, src, dst, nsum, n_edges);
    }
    k_sage_gemm<<<gblk, gthr, 0, stream>>>(h1, nsum, inv, w_self2, w_neigh2, b2,
                                           out, n_nodes, /*relu=*/0);
}